// RowPool_74680891343504
// MI455X (gfx1250) — compile-verified
//
#include <hip/hip_runtime.h>
#include <cstdint>
#include <cstddef>

// RowPool masked cummax: Y[n,c,t,:] = cummax(where(mask[n,:], -inf, X[n,c,t,:]))
// Shapes: X (4,128,256,512) f32, mask (4,512) int (bool), Y same as X.
// Memory-bound: 512MB traffic -> ~22us floor @ 23.3 TB/s on MI455X.

#define TS     512                 // row length (Ts)
#define CHUNK  16                  // contiguous elements per lane (512/32)
#define PADW   20                  // LDS words per 16-word chunk: 16B-aligned, low bank conflicts
#define WAVE_LDS_WORDS (32 * PADW) // 640 words = 2560 B per wave
#define WPB    8                   // waves per block (256 threads)

#if defined(__has_builtin)
#if __has_builtin(__builtin_amdgcn_global_load_async_to_lds_b128) && \
    __has_builtin(__builtin_amdgcn_s_wait_asynccnt)
#define RP_ASYNC 1
#endif
#endif
#ifndef RP_ASYNC
#define RP_ASYNC 0
#endif

#if RP_ASYNC
// Builtin signature (from hipcc diagnostic): param 1 is a pointer to a
// 16-byte int vector in the global (AS1, printed "__device__") address space.
typedef int v4i __attribute__((vector_size(16)));
typedef __attribute__((address_space(1))) v4i* gv4i_p;   // global int4*
typedef __attribute__((address_space(3))) v4i* lv4i_p;   // LDS int4*
#endif

__global__ __launch_bounds__(WPB * 32)
void rowpool_cummax_kernel(const float* __restrict__ X,
                           const int*   __restrict__ srcMask,
                           float*       __restrict__ Y,
                           int nRows, int rowsPerN) {
  __shared__ float lds[WPB * WAVE_LDS_WORDS];

  const int tid  = threadIdx.x;
  const int wave = tid >> 5;
  const int lane = tid & 31;
  const int row  = blockIdx.x * WPB + wave;   // one wave32 per 512-float row
  if (row >= nRows) return;

  const int n = row / rowsPerN;               // batch index for mask
  const float* rowX = X + (size_t)row * TS;
  float*       rowY = Y + (size_t)row * TS;
  float*       wlds = &lds[wave * WAVE_LDS_WORDS];

  const float NEG_INF = -__builtin_inff();
  const int   sbase   = lane * CHUNK;

  float v[CHUNK];

#if RP_ASYNC
  // Stage row into LDS via async DMA: each b128 instruction moves a fully
  // contiguous 512B burst (32 lanes x 16B). LDS destination uses padded
  // layout: chunk c starts at word PADW*c (PADW*4 bytes is 16B aligned).
  #pragma unroll
  for (int k = 0; k < 4; ++k) {
    const int s0 = 4 * (lane + 32 * k);              // multiple of 4
    const int w  = PADW * (s0 >> 4) + (s0 & 15);     // padded word index
    __builtin_amdgcn_global_load_async_to_lds_b128(
        (gv4i_p)(rowX + s0),
        (lv4i_p)(wlds + w),
        0, 0);
  }
  __builtin_amdgcn_s_wait_asynccnt(0);
  __asm__ __volatile__("" ::: "memory");
  // Lane l owns chunk l: words [PADW*l, PADW*l+16). Stride-20 start banks
  // give at worst 2-way conflicts on these b32 reads.
  #pragma unroll
  for (int i = 0; i < CHUNK; ++i) v[i] = wlds[PADW * lane + i];
#else
  // Fallback: direct strided float4 loads (L0/L2 reassemble full lines).
  const float4* rx4 = (const float4*)(rowX + sbase);
  #pragma unroll
  for (int j = 0; j < 4; ++j) {
    float4 t = rx4[j];
    v[4*j+0] = t.x; v[4*j+1] = t.y; v[4*j+2] = t.z; v[4*j+3] = t.w;
  }
#endif

  // Apply source mask (True/nonzero => -inf). Mask row is tiny and L2-hot.
  const int4* m4 = (const int4*)(srcMask + n * TS + sbase);
  #pragma unroll
  for (int j = 0; j < 4; ++j) {
    int4 m = m4[j];
    if (m.x) v[4*j+0] = NEG_INF;
    if (m.y) v[4*j+1] = NEG_INF;
    if (m.z) v[4*j+2] = NEG_INF;
    if (m.w) v[4*j+3] = NEG_INF;
  }

  // 1) serial inclusive max-scan over this lane's 16 contiguous elements
  #pragma unroll
  for (int i = 1; i < CHUNK; ++i) v[i] = fmaxf(v[i], v[i-1]);

  // 2) wave-level inclusive max-scan of lane totals (5 shuffle steps, wave32)
  float inc = v[CHUNK - 1];
  #pragma unroll
  for (int d = 1; d < 32; d <<= 1) {
    float t = __shfl_up(inc, (unsigned)d, 32);
    if (lane >= d) inc = fmaxf(inc, t);
  }
  // exclusive prefix for this lane
  float pre = __shfl_up(inc, 1u, 32);
  if (lane == 0) pre = NEG_INF;

  // 3) fold exclusive prefix into local scan
  #pragma unroll
  for (int i = 0; i < CHUNK; ++i) v[i] = fmaxf(v[i], pre);

  // Write back as float4 (wave covers the whole 2KB row; L2 write-combines).
  float4* ry4 = (float4*)(rowY + sbase);
  #pragma unroll
  for (int j = 0; j < 4; ++j) {
    float4 o;
    o.x = v[4*j+0]; o.y = v[4*j+1]; o.z = v[4*j+2]; o.w = v[4*j+3];
    ry4[j] = o;
  }
}

extern "C" void kernel_launch(void* const* d_in, const int* in_sizes, int n_in,
                              void* d_out, int out_size, void* d_ws, size_t ws_size,
                              hipStream_t stream) {
  (void)n_in; (void)out_size; (void)d_ws; (void)ws_size;
  const float* X    = (const float*)d_in[0];
  const int*   mask = (const int*)d_in[1];     // bool mask delivered as int
  float*       Y    = (float*)d_out;

  const int total    = in_sizes[0];            // N*C*Tt*Ts
  const int nRows    = total / TS;             // N*C*Tt = 131072
  const int N        = in_sizes[1] / TS;       // mask is [N, Ts]
  const int rowsPerN = nRows / N;              // C*Tt

  dim3 block(WPB * 32);
  dim3 grid((nRows + WPB - 1) / WPB);
  rowpool_cummax_kernel<<<grid, block, 0, stream>>>(X, mask, Y, nRows, rowsPerN);
}